// PathScorer_54374285968081
// MI455X (gfx1250) — compile-verified
//
#include <hip/hip_runtime.h>
#include <stdint.h>
#include <math.h>

#define NDIM 64
#define R2   193
#define L3   3

typedef uint32_t u32x4 __attribute__((ext_vector_type(4)));
typedef int      i32x4 __attribute__((ext_vector_type(4)));
typedef int      i32x8 __attribute__((ext_vector_type(8)));

#if defined(__has_builtin)
#  if __has_builtin(__builtin_amdgcn_tensor_load_to_lds)
#    define HAS_TDM 1
#  endif
#  if __has_builtin(__builtin_amdgcn_global_load_async_to_lds_b32)
#    define HAS_ASYNC_LDS 1
#  endif
#  if __has_builtin(__builtin_amdgcn_s_wait_tensorcnt)
#    define HAS_WAIT_TENSOR_BUILTIN 1
#  endif
#  if __has_builtin(__builtin_amdgcn_s_wait_asynccnt)
#    define HAS_WAIT_ASYNC_BUILTIN 1
#  endif
#endif

typedef __attribute__((address_space(1))) int gint_t;  // global
typedef __attribute__((address_space(3))) int lint_t;  // LDS

__device__ __forceinline__ void wait_tensorcnt_zero() {
#if defined(HAS_WAIT_TENSOR_BUILTIN)
  __builtin_amdgcn_s_wait_tensorcnt(0);
#else
  asm volatile("s_wait_tensorcnt 0x0" ::: "memory");
#endif
}

__device__ __forceinline__ void wait_asynccnt_zero() {
#if defined(HAS_WAIT_ASYNC_BUILTIN)
  __builtin_amdgcn_s_wait_asynccnt(0);
#else
  asm volatile("s_wait_asynccnt 0" ::: "memory");
#endif
}

#if defined(HAS_TDM)
// TDM: DMA one gathered 64x64 f32 matrix (column r of the 193-wide last dim,
// i.e. a 4096-row x 1-elem tile with row stride 193) from global into LDS.
__device__ __forceinline__ void tdm_gather_col(const float* gsrc, uint32_t lds_off) {
  const uint64_t ga = (uint64_t)(uintptr_t)gsrc;

  u32x4 g0;
  g0[0] = 1u;                                   // count=1 (valid), user-mode, no gather
  g0[1] = lds_off;                              // lds_addr (byte offset in LDS)
  g0[2] = (uint32_t)ga;                         // global_addr[31:0]
  g0[3] = (uint32_t)((ga >> 32) & 0x01FFFFFFu)  // global_addr[56:32]
          | (2u << 30);                         // type = 2 ("image")

  i32x8 g1;
  g1[0] = (int)(2u << 16);          // workgroup_mask=0, data_size=2 (4B), no flags
  g1[1] = (int)(1u << 16);          // atomic_barrier_addr=0 ; tensor_dim0[15:0]=1
  g1[2] = (int)(4096u << 16);       // tensor_dim0[31:16]=0 ; tensor_dim1[15:0]=4096
  g1[3] = (int)(1u << 16);          // tensor_dim1[31:16]=0 ; tile_dim0=1
  g1[4] = (int)4096;                // tile_dim1=4096 ; tile_dim2=0
  g1[5] = (int)R2;                  // tensor_dim0_stride[31:0] = 193
  g1[6] = 0;                        // stride0[47:32]=0 ; tensor_dim1_stride[15:0]=0
  g1[7] = 0;                        // tensor_dim1_stride[47:16]=0

  i32x4 g2 = {0, 0, 0, 0};          // dims 2/3 unused (tile_dim3=0)
  i32x4 g3 = {0, 0, 0, 0};          // dims 3/4 unused

#if __clang_major__ >= 23
  i32x8 g4 = {0, 0, 0, 0, 0, 0, 0, 0};
  __builtin_amdgcn_tensor_load_to_lds(g0, g1, g2, g3, g4, 0);
#else
  __builtin_amdgcn_tensor_load_to_lds(g0, g1, g2, g3, 0);
#endif
}
#endif  // HAS_TDM

// acc[ii][kk] = max_j ( A[i0+ii][j] + B[j][k0+kk] ) over 64x64 LDS tiles.
// Clean 64-iteration loop (16 x unroll-4, no peel/remainder): init to -inf.
__device__ __forceinline__ void maxplus_tile(const float* __restrict__ A,
                                             const float* __restrict__ Bm,
                                             float acc[4][4], int i0, int k0) {
#pragma unroll
  for (int ii = 0; ii < 4; ++ii)
#pragma unroll
    for (int kk = 0; kk < 4; ++kk) acc[ii][kk] = -INFINITY;

#pragma unroll 4
  for (int j = 0; j < NDIM; ++j) {
    const float4 bv = *(const float4*)(Bm + j * NDIM + k0);
#pragma unroll
    for (int ii = 0; ii < 4; ++ii) {
      const float a = A[(i0 + ii) * NDIM + j];
      acc[ii][0] = fmaxf(acc[ii][0], a + bv.x);
      acc[ii][1] = fmaxf(acc[ii][1], a + bv.y);
      acc[ii][2] = fmaxf(acc[ii][2], a + bv.z);
      acc[ii][3] = fmaxf(acc[ii][3], a + bv.w);
    }
  }
}

__global__ __launch_bounds__(256) void PathScorer_54374285968081_kernel(
    const float* __restrict__ trans,   // [B, 64, 64, 193]
    const int*   __restrict__ chains,  // [Nc, 3]
    float*       __restrict__ out,     // [B, 64, 64, Nc]
    int Nc) {
  __shared__ __align__(16) float s0[NDIM * NDIM];  // reused as the stage-1 result T
  __shared__ __align__(16) float s1[NDIM * NDIM];
  __shared__ __align__(16) float s2[NDIM * NDIM];

  const int c = blockIdx.x;   // chain   (2D grid: no integer division)
  const int b = blockIdx.y;   // batch

  const int tid = threadIdx.x;
  const float* tb = trans + (size_t)b * (NDIM * NDIM * R2);

  const int r0 = __builtin_amdgcn_readfirstlane(chains[c * L3 + 0]);
  const int r1 = __builtin_amdgcn_readfirstlane(chains[c * L3 + 1]);
  const int r2 = __builtin_amdgcn_readfirstlane(chains[c * L3 + 2]);

  // ---- gather the three 64x64 matrices (stride-193 columns) into LDS ----
#if defined(HAS_TDM)
  if (tid < 32) {  // one wave issues the three TDM descriptors (TDM ignores EXEC)
    tdm_gather_col(tb + r0, (uint32_t)(uintptr_t)(lint_t*)&s0[0]);
    tdm_gather_col(tb + r1, (uint32_t)(uintptr_t)(lint_t*)&s1[0]);
    tdm_gather_col(tb + r2, (uint32_t)(uintptr_t)(lint_t*)&s2[0]);
    wait_tensorcnt_zero();
  }
#elif defined(HAS_ASYNC_LDS)
#pragma unroll
  for (int q = 0; q < 16; ++q) {
    const int e   = tid + q * 256;   // flattened (i*64 + j)
    const int off = e * R2;
    __builtin_amdgcn_global_load_async_to_lds_b32(
        (gint_t*)(tb + off + r0), (lint_t*)&s0[e], 0, 0);
    __builtin_amdgcn_global_load_async_to_lds_b32(
        (gint_t*)(tb + off + r1), (lint_t*)&s1[e], 0, 0);
    __builtin_amdgcn_global_load_async_to_lds_b32(
        (gint_t*)(tb + off + r2), (lint_t*)&s2[e], 0, 0);
  }
  wait_asynccnt_zero();
#else
  for (int q = 0; q < 16; ++q) {
    const int e   = tid + q * 256;
    const int off = e * R2;
    s0[e] = tb[off + r0];
    s1[e] = tb[off + r1];
    s2[e] = tb[off + r2];
  }
#endif
  __syncthreads();

  // 16x16 lane grid, 4x4 register tile per lane
  const int tx = tid & 15;
  const int ty = tid >> 4;
  const int i0 = ty * 4;
  const int k0 = tx * 4;

  // ---- stage 1: T = maxplus(s0, s1) ----
  float acc[4][4];
  maxplus_tile(s0, s1, acc, i0, k0);

  __syncthreads();  // everyone done reading s0
#pragma unroll
  for (int ii = 0; ii < 4; ++ii) {
    float4 v;
    v.x = acc[ii][0]; v.y = acc[ii][1]; v.z = acc[ii][2]; v.w = acc[ii][3];
    *(float4*)&s0[(i0 + ii) * NDIM + k0] = v;  // T overwrites s0
  }
  __syncthreads();

  // ---- stage 2: scores = maxplus(T, s2) ----
  float acc2[4][4];
  maxplus_tile(s0, s2, acc2, i0, k0);

  // scatter store: out[((b*64 + i)*64 + k)*Nc + c]
#pragma unroll
  for (int ii = 0; ii < 4; ++ii) {
#pragma unroll
    for (int kk = 0; kk < 4; ++kk) {
      const size_t oidx =
          ((((size_t)b * NDIM) + (i0 + ii)) * NDIM + (k0 + kk)) * (size_t)Nc + c;
      out[oidx] = acc2[ii][kk];
    }
  }
}

extern "C" void kernel_launch(void* const* d_in, const int* in_sizes, int n_in,
                              void* d_out, int out_size, void* d_ws, size_t ws_size,
                              hipStream_t stream) {
  const float* trans  = (const float*)d_in[0];
  const int*   chains = (const int*)d_in[1];
  float*       out    = (float*)d_out;

  const int B  = in_sizes[0] / (NDIM * NDIM * R2);  // = 8
  const int Nc = in_sizes[1] / L3;                  // = 128

  dim3 grid(Nc, B);    // (chain, batch) — no integer division in-kernel
  dim3 block(256);     // 8 waves (wave32), 16x16 lane grid, 4x4 tiles
  PathScorer_54374285968081_kernel<<<grid, block, 0, stream>>>(trans, chains, out, Nc);
}